// Scalar3DHMM_63952063037752
// MI455X (gfx1250) — compile-verified
//
#include <hip/hip_runtime.h>
#include <math.h>

// Problem constants (match reference)
#define NSTATE 2048   // N = 32*32*2
#define NV     10000  // vocab
#define NB     4      // batch (stories)
#define NT     16     // timesteps
#define NL     16     // tokens per timestep

typedef float v2f __attribute__((ext_vector_type(2)));
typedef float v8f __attribute__((ext_vector_type(8)));

// ---------------------------------------------------------------------------
// K0: transition log-softmax over the 7 neighbor slots.  [N,7] -> [N,7]
// ---------------------------------------------------------------------------
__global__ __launch_bounds__(256) void k_trans_ls(const float* __restrict__ tw,
                                                  float* __restrict__ out) {
  int r = blockIdx.x * 256 + threadIdx.x;
  if (r >= NSTATE) return;
  float v[7];
  float m = -INFINITY;
#pragma unroll
  for (int j = 0; j < 7; ++j) { v[j] = tw[r * 7 + j]; m = fmaxf(m, v[j]); }
  float s = 0.f;
#pragma unroll
  for (int j = 0; j < 7; ++j) s += __expf(v[j] - m);
  float l = m + __logf(s);
#pragma unroll
  for (int j = 0; j < 7; ++j) out[r * 7 + j] = v[j] - l;
}

// ---------------------------------------------------------------------------
// K1: prior log-softmax over N.  [N] -> [N]   (single block)
// ---------------------------------------------------------------------------
__global__ __launch_bounds__(256) void k_prior_ls(const float* __restrict__ pw,
                                                  float* __restrict__ out) {
  __shared__ float red[256];
  int tid = threadIdx.x;
  float m = -INFINITY;
  for (int i = tid; i < NSTATE; i += 256) m = fmaxf(m, pw[i]);
  red[tid] = m; __syncthreads();
  for (int off = 128; off > 0; off >>= 1) {
    if (tid < off) red[tid] = fmaxf(red[tid], red[tid + off]);
    __syncthreads();
  }
  float M = red[0]; __syncthreads();
  float s = 0.f;
  for (int i = tid; i < NSTATE; i += 256) s += __expf(pw[i] - M);
  red[tid] = s; __syncthreads();
  for (int off = 128; off > 0; off >>= 1) {
    if (tid < off) red[tid] += red[tid + off];
    __syncthreads();
  }
  float lse = M + __logf(red[0]);
  for (int i = tid; i < NSTATE; i += 256) out[i] = pw[i] - lse;
}

// ---------------------------------------------------------------------------
// K2: per-row logsumexp of emission_w over V.  One block per row, two passes
// (max, then sum-exp; pass 2 hits L2/WGP$).  float4 coalesced + prefetch.
// ---------------------------------------------------------------------------
__global__ __launch_bounds__(256) void k_emis_lse(const float* __restrict__ E,
                                                  float* __restrict__ lse) {
  const int row = blockIdx.x;
  const int tid = threadIdx.x;
  const float4* rp4 = (const float4*)(E + (size_t)row * NV);  // 40000B rows: 16B aligned
  const int nvec = NV / 4;  // 2500 exactly

  __shared__ float red[256];

  float m = -INFINITY;
  for (int i = tid; i < nvec; i += 256) {
    __builtin_prefetch(rp4 + i + 256, 0, 1);   // global_prefetch_b8 (speculative OK)
    float4 x = rp4[i];
    m = fmaxf(m, fmaxf(fmaxf(x.x, x.y), fmaxf(x.z, x.w)));
  }
  red[tid] = m; __syncthreads();
  for (int off = 128; off > 0; off >>= 1) {
    if (tid < off) red[tid] = fmaxf(red[tid], red[tid + off]);
    __syncthreads();
  }
  float M = red[0]; __syncthreads();

  float s = 0.f;
  for (int i = tid; i < nvec; i += 256) {
    float4 x = rp4[i];
    s += __expf(x.x - M) + __expf(x.y - M) + __expf(x.z - M) + __expf(x.w - M);
  }
  red[tid] = s; __syncthreads();
  for (int off = 128; off > 0; off >>= 1) {
    if (tid < off) red[tid] += red[tid + off];
    __syncthreads();
  }
  if (tid == 0) lse[row] = M + __logf(red[0]);
}

// ---------------------------------------------------------------------------
// K3: emission token sums via WMMA.
//   emisSum[t,b,n] = sum_l emission_w[n, tok[b,t,l]] - 16*lse[n]
// One block per (t,b); 8 waves; each wave reduces 16-state tiles.
// The 16-value reduction runs on the matrix unit:
//   D = ones(16x4) x Gathered(4x16) + C accumulated over 4 K-chunks
// A layout (16x4 f32): lanes 0-15 hold M=0..15 {K=0,K=1}; lanes 16-31 {K=2,K=3}.
// B layout (4x16 f32): lanes 0-15 hold N=0..15 {K=0,K=1}; lanes 16-31 {K=2,K=3}.
// Every row of D holds the per-column sum; lane l<16 reads acc[0] for n=base+l.
// ---------------------------------------------------------------------------
__global__ __launch_bounds__(256) void k_emis_tok_wmma(const float* __restrict__ E,
                                                       const float* __restrict__ lse,
                                                       const int* __restrict__ stories,
                                                       float* __restrict__ emisSum) {
  const int tb = blockIdx.x;      // tb = t*NB + b
  const int t  = tb >> 2;         // NB == 4
  const int b  = tb & 3;

  __shared__ int toks[NL];
  if (threadIdx.x < NL) toks[threadIdx.x] = stories[(b * NT + t) * NL + threadIdx.x];
  __syncthreads();

  const int wave = threadIdx.x >> 5;   // 0..7
  const int lane = threadIdx.x & 31;
  const int nl   = lane & 15;          // N index within tile
  const int kb   = (lane >> 4) * 2;    // K base: 0 or 2

  v2f a; a.x = 1.0f; a.y = 1.0f;       // ones A-matrix -> column sums

  for (int tile = wave; tile < NSTATE / 16; tile += 8) {   // uniform per wave
    const int n = tile * 16 + nl;
    const float* rowp = E + (size_t)n * NV;
    v8f acc = {};
#pragma unroll
    for (int c = 0; c < 4; ++c) {      // 4 K-chunks x K=4 = 16 tokens
      v2f bv;
      bv.x = rowp[toks[4 * c + kb + 0]];
      bv.y = rowp[toks[4 * c + kb + 1]];
      acc = __builtin_amdgcn_wmma_f32_16x16x4_f32(
          /*neg_a=*/false, a, /*neg_b=*/false, bv,
          /*c_mod=*/(short)0, acc, /*reuse_a=*/false, /*reuse_b=*/false);
    }
    if (lane < 16)
      emisSum[((size_t)tb << 11) + n] = acc[0] - 16.0f * lse[n];
  }
}

// ---------------------------------------------------------------------------
// K4: forward recursion.  One block per story b (chains independent over b).
// alpha lives in LDS (8 KB); 16 sequential steps with block barriers.
// Band taps (scatter-last-wins: delta 2048 == delta 0 mod N, slot 6 survives):
//   j=1: i=h-1   j=2: i=h+1   j=3: i=h-32   j=4: i=h+32   j=5: i=h-1024   j=6: i=h
// ---------------------------------------------------------------------------
__global__ __launch_bounds__(256) void k_forward(const float* __restrict__ emisSum,
                                                 const float* __restrict__ trans7,
                                                 const float* __restrict__ priors,
                                                 float* __restrict__ out) {
  const int b = blockIdx.x;
  const int tid = threadIdx.x;
  __shared__ float alpha[NSTATE];

  // t = 0 : alpha0 = emisSum[0,b,:] + priors
  for (int h = tid; h < NSTATE; h += 256) {
    float v = emisSum[(size_t)(0 * NB + b) * NSTATE + h] + priors[h];
    alpha[h] = v;
    out[(size_t)(0 * NB + b) * NSTATE + h] = v;
  }
  __syncthreads();

  for (int t = 1; t < NT; ++t) {
    float nv[NSTATE / 256];
#pragma unroll
    for (int k = 0; k < NSTATE / 256; ++k) {
      const int h = tid + k * 256;
      const float* tr = trans7 + h * 7;
      float term[6];
      term[0] = tr[1] + alpha[(h - 1)    & (NSTATE - 1)];
      term[1] = tr[2] + alpha[(h + 1)    & (NSTATE - 1)];
      term[2] = tr[3] + alpha[(h - 32)   & (NSTATE - 1)];
      term[3] = tr[4] + alpha[(h + 32)   & (NSTATE - 1)];
      term[4] = tr[5] + alpha[(h - 1024) & (NSTATE - 1)];
      term[5] = tr[6] + alpha[h];
      float m = term[0];
#pragma unroll
      for (int j = 1; j < 6; ++j) m = fmaxf(m, term[j]);
      float s = 0.f;
#pragma unroll
      for (int j = 0; j < 6; ++j) s += __expf(term[j] - m);
      nv[k] = emisSum[(size_t)(t * NB + b) * NSTATE + h] + m + __logf(s);
    }
    __syncthreads();
#pragma unroll
    for (int k = 0; k < NSTATE / 256; ++k) {
      const int h = tid + k * 256;
      alpha[h] = nv[k];
      out[(size_t)(t * NB + b) * NSTATE + h] = nv[k];
    }
    __syncthreads();
  }
}

// ---------------------------------------------------------------------------
// Launch
// ---------------------------------------------------------------------------
extern "C" void kernel_launch(void* const* d_in, const int* in_sizes, int n_in,
                              void* d_out, int out_size, void* d_ws, size_t ws_size,
                              hipStream_t stream) {
  const int*   stories = (const int*)d_in[0];      // [B,T,L] int32
  // d_in[1] = story_length (scalar 16, compile-time here)
  const float* tw      = (const float*)d_in[2];    // [N,7]
  const float* ew      = (const float*)d_in[3];    // [N,V]
  const float* pw      = (const float*)d_in[4];    // [N]
  float*       out     = (float*)d_out;            // [T,B,N]

  float* ws       = (float*)d_ws;
  float* trans7n  = ws;                            // N*7      = 14336 floats
  float* priorsn  = ws + NSTATE * 7;               // N        =  2048
  float* lse      = priorsn + NSTATE;              // N        =  2048
  float* emisSum  = lse + NSTATE;                  // T*B*N    = 131072

  k_trans_ls     <<<(NSTATE + 255) / 256, 256, 0, stream>>>(tw, trans7n);
  k_prior_ls     <<<1,                    256, 0, stream>>>(pw, priorsn);
  k_emis_lse     <<<NSTATE,               256, 0, stream>>>(ew, lse);
  k_emis_tok_wmma<<<NT * NB,              256, 0, stream>>>(ew, lse, stories, emisSum);
  k_forward      <<<NB,                   256, 0, stream>>>(emisSum, trans7n, priorsn, out);
}